// GlobalModulatedGNN_88510686036742
// MI455X (gfx1250) — compile-verified
//
#include <hip/hip_runtime.h>
#include <hip/hip_bf16.h>
#include <math.h>

#define GAT_HEADS 4
#define GAT_OUT 32
#define GAT_IN 128
#define GAT_HC (GAT_HEADS * GAT_OUT)   // 128
#define GAT_GDIM 64
#define GAT_NEG 0.2f

typedef __attribute__((ext_vector_type(2))) float v2f;
typedef __attribute__((ext_vector_type(8))) float v8f;

// ---------------------------------------------------------------------------
// Portable float atomic-max via int max / uint min bit ordering.
// Works for mixed-sign values (IEEE-754 total order trick).
// ---------------------------------------------------------------------------
__device__ __forceinline__ void atomicMaxF32(float* addr, float val) {
    if (val >= 0.0f) {
        atomicMax(reinterpret_cast<int*>(addr), __float_as_int(val));
    } else {
        atomicMin(reinterpret_cast<unsigned int*>(addr), __float_as_uint(val));
    }
}

__device__ __forceinline__ float lrelu(float v) {
    return v > 0.0f ? v : GAT_NEG * v;
}

// ---------------------------------------------------------------------------
// Workspace init
// ---------------------------------------------------------------------------
__global__ void fill_kernel(float* __restrict__ p, float v, long long n) {
    long long i = (long long)blockIdx.x * blockDim.x + threadIdx.x;
    if (i < n) p[i] = v;
}

// ---------------------------------------------------------------------------
// xh = x @ W  via V_WMMA_F32_16X16X4_F32.
// Block = 256 threads = 8 waves. Block b computes rows [16b,16b+16);
// wave w computes cols [16w, 16w+16). K-loop: 128/4 = 32 WMMA issues.
// Per-lane operand layout (wave32):
//   A 16x4 : lane l holds A[l%16][2*(l/16) + {0,1}]  (contiguous float2)
//   B 4x16 : lane l holds B[2*(l/16) + {0,1}][l%16]
//   C/D    : vgpr r, lane l -> D[r + 8*(l/16)][l%16]
// ---------------------------------------------------------------------------
__global__ __launch_bounds__(256)
void gemm_xw_wmma_kernel(const float* __restrict__ x,
                         const float* __restrict__ W,
                         float* __restrict__ xh, int n) {
    const int wave = threadIdx.x >> 5;
    const int lane = threadIdx.x & 31;
    const int m    = lane & 15;
    const int grp  = lane >> 4;          // 0 or 1
    const int row0 = blockIdx.x * 16;
    const int col0 = wave * 16;
    if (row0 + m >= n) return;           // N is a multiple of 16; guard anyway

    const float* __restrict__ arow = x + (long long)(row0 + m) * GAT_IN;
    v8f acc = {};
#pragma unroll
    for (int kc = 0; kc < GAT_IN; kc += 4) {
        const int kk = kc + 2 * grp;
        v2f a = *reinterpret_cast<const v2f*>(arow + kk);   // 8B aligned (kk even)
        v2f b;
        b.x = W[(long long)kk * GAT_HC + col0 + m];
        b.y = W[(long long)(kk + 1) * GAT_HC + col0 + m];
        acc = __builtin_amdgcn_wmma_f32_16x16x4_f32(
            /*neg_a=*/false, a, /*neg_b=*/false, b,
            /*c_mod=*/(short)0, acc, /*reuse_a=*/false, /*reuse_b=*/false);
    }
#pragma unroll
    for (int r = 0; r < 8; ++r) {
        const int row = row0 + r + 8 * grp;
        xh[(long long)row * GAT_HC + col0 + m] = acc[r];
    }
}

// ---------------------------------------------------------------------------
// a_src[n][h] = <xh[n,h,:], att_src[h,:]>, same for a_dst. One thread per (n,h).
// ---------------------------------------------------------------------------
__global__ void attn_scores_kernel(const float* __restrict__ xh,
                                   const float* __restrict__ att_src,
                                   const float* __restrict__ att_dst,
                                   float* __restrict__ a_src,
                                   float* __restrict__ a_dst, int n) {
    const int idx = blockIdx.x * blockDim.x + threadIdx.x;
    if (idx >= n * GAT_HEADS) return;
    const int node = idx >> 2;
    const int h    = idx & 3;
    const float* __restrict__ row = xh + (long long)node * GAT_HC + h * GAT_OUT;
    const float* __restrict__ as  = att_src + h * GAT_OUT;
    const float* __restrict__ ad  = att_dst + h * GAT_OUT;
    float ss = 0.0f, sd = 0.0f;
#pragma unroll
    for (int c = 0; c < GAT_OUT; ++c) {
        const float v = row[c];
        ss = fmaf(v, as[c], ss);
        sd = fmaf(v, ad[c], sd);
    }
    a_src[idx] = ss;
    a_dst[idx] = sd;
}

// ---------------------------------------------------------------------------
// Pass 1 over edges (incl. self-loops): alpha = leakyrelu(a_src[s]+a_dst[d]);
// store alpha, atomic-max into amax[dst].
// ---------------------------------------------------------------------------
__global__ void edge_alpha_max_kernel(const int* __restrict__ src,
                                      const int* __restrict__ dst,
                                      const float* __restrict__ a_src,
                                      const float* __restrict__ a_dst,
                                      float* __restrict__ alpha,
                                      float* __restrict__ amax,
                                      int nE, int n) {
    const int e = blockIdx.x * blockDim.x + threadIdx.x;
    const int tot = nE + n;
    if (e >= tot) return;
    int s, d;
    if (e < nE) { s = src[e]; d = dst[e]; } else { s = d = e - nE; }
    const float4 as = *reinterpret_cast<const float4*>(a_src + (long long)s * 4);
    const float4 ad = *reinterpret_cast<const float4*>(a_dst + (long long)d * 4);
    float4 al;
    al.x = lrelu(as.x + ad.x);
    al.y = lrelu(as.y + ad.y);
    al.z = lrelu(as.z + ad.z);
    al.w = lrelu(as.w + ad.w);
    *reinterpret_cast<float4*>(alpha + (long long)e * 4) = al;
    float* amx = amax + (long long)d * 4;
    atomicMaxF32(amx + 0, al.x);
    atomicMaxF32(amx + 1, al.y);
    atomicMaxF32(amx + 2, al.z);
    atomicMaxF32(amx + 3, al.w);
}

// ---------------------------------------------------------------------------
// Pass 2: ex = exp(alpha - amax[dst]); store ex in-place; atomic-add denom[dst].
// ---------------------------------------------------------------------------
__global__ void edge_exp_sum_kernel(const int* __restrict__ src,
                                    const int* __restrict__ dst,
                                    float* __restrict__ alpha,   // in: alpha, out: ex
                                    const float* __restrict__ amax,
                                    float* __restrict__ denom,
                                    int nE, int n) {
    const int e = blockIdx.x * blockDim.x + threadIdx.x;
    const int tot = nE + n;
    if (e >= tot) return;
    int d;
    if (e < nE) { d = dst[e]; } else { d = e - nE; }
    float4 al = *reinterpret_cast<const float4*>(alpha + (long long)e * 4);
    const float4 mx = *reinterpret_cast<const float4*>(amax + (long long)d * 4);
    float4 ex;
    ex.x = __expf(al.x - mx.x);
    ex.y = __expf(al.y - mx.y);
    ex.z = __expf(al.z - mx.z);
    ex.w = __expf(al.w - mx.w);
    *reinterpret_cast<float4*>(alpha + (long long)e * 4) = ex;
    float* dn = denom + (long long)d * 4;
    atomicAdd(dn + 0, ex.x);
    atomicAdd(dn + 1, ex.y);
    atomicAdd(dn + 2, ex.z);
    atomicAdd(dn + 3, ex.w);
}

// ---------------------------------------------------------------------------
// Pass 3 (dominant, HBM/atomic bound): for each edge, all 128 channels:
// agg[dst][t] += (ex[e][h]/(denom[dst][h]+eps)) * xh[src][t],  h = t/32.
// 256-thread block handles 2 edges; head index is uniform per wave,
// so ex/denom loads broadcast.
// ---------------------------------------------------------------------------
__global__ __launch_bounds__(256)
void edge_scatter_kernel(const int* __restrict__ src,
                         const int* __restrict__ dst,
                         const float* __restrict__ xh,
                         const float* __restrict__ ex,
                         const float* __restrict__ denom,
                         float* __restrict__ agg,
                         int nE, int n) {
    const int e = blockIdx.x * 2 + (threadIdx.x >> 7);
    const int tot = nE + n;
    if (e >= tot) return;
    const int t = threadIdx.x & 127;
    const int h = t >> 5;              // uniform within a wave
    int s, d;
    if (e < nE) { s = src[e]; d = dst[e]; } else { s = d = e - nE; }
    const float coef = ex[(long long)e * 4 + h] /
                       (denom[(long long)d * 4 + h] + 1e-16f);
    atomicAdd(agg + (long long)d * GAT_HC + t,
              coef * xh[(long long)s * GAT_HC + t]);
}

// ---------------------------------------------------------------------------
// FiLM modulators: fm[c] = tanh(g@Wf + bf)[c], bm[c] = (g@Wb + bb)[c].
// ---------------------------------------------------------------------------
__global__ void film_kernel(const float* __restrict__ g,
                            const float* __restrict__ Wf,
                            const float* __restrict__ bf,
                            const float* __restrict__ Wb,
                            const float* __restrict__ bb,
                            float* __restrict__ fmbm) {
    const int c = threadIdx.x;
    if (c >= GAT_OUT) return;
    float sf = 0.0f, sb = 0.0f;
#pragma unroll
    for (int k = 0; k < GAT_GDIM; ++k) {
        const float gv = g[k];
        sf = fmaf(gv, Wf[k * GAT_OUT + c], sf);
        sb = fmaf(gv, Wb[k * GAT_OUT + c], sb);
    }
    fmbm[c]           = tanhf(sf + bf[c]);
    fmbm[GAT_OUT + c] = sb + bb[c];
}

// ---------------------------------------------------------------------------
// Finalize: head mean + gat_bias, then FiLM modulation.
// ---------------------------------------------------------------------------
__global__ void finalize_kernel(const float* __restrict__ agg,
                                const float* __restrict__ gat_bias,
                                const float* __restrict__ fmbm,
                                float* __restrict__ out, int n) {
    const int idx = blockIdx.x * blockDim.x + threadIdx.x;
    if (idx >= n * GAT_OUT) return;
    const int node = idx >> 5;
    const int c    = idx & 31;
    const float* __restrict__ a = agg + (long long)node * GAT_HC;
    const float mean = 0.25f * (a[c] + a[GAT_OUT + c] +
                                a[2 * GAT_OUT + c] + a[3 * GAT_OUT + c]);
    const float v = mean + gat_bias[c];
    out[idx] = v * (1.0f + fmbm[c]) + fmbm[GAT_OUT + c];
}

// ---------------------------------------------------------------------------
extern "C" void kernel_launch(void* const* d_in, const int* in_sizes, int n_in,
                              void* d_out, int out_size, void* d_ws, size_t ws_size,
                              hipStream_t stream) {
    const float* x        = (const float*)d_in[0];
    const int*   ei       = (const int*)d_in[1];   // [2,E] int32 (jax x64 off)
    const float* g        = (const float*)d_in[2];
    const float* W        = (const float*)d_in[3];
    const float* att_src  = (const float*)d_in[4];
    const float* att_dst  = (const float*)d_in[5];
    const float* gat_bias = (const float*)d_in[6];
    const float* Wf       = (const float*)d_in[7];
    const float* bf       = (const float*)d_in[8];
    const float* Wb       = (const float*)d_in[9];
    const float* bb       = (const float*)d_in[10];

    const int n  = in_sizes[0] / GAT_IN;   // 50000
    const int nE = in_sizes[1] / 2;        // 1600000
    const int tot = nE + n;
    const int* src = ei;
    const int* dst = ei + nE;

    // Workspace layout (floats)
    float* ws = (float*)d_ws;
    size_t off = 0;
    float* xh    = ws + off; off += (size_t)n * GAT_HC;    // 25.6 MB
    float* a_src = ws + off; off += (size_t)n * GAT_HEADS;
    float* a_dst = ws + off; off += (size_t)n * GAT_HEADS;
    float* amax  = ws + off; off += (size_t)n * GAT_HEADS;
    float* denom = ws + off; off += (size_t)n * GAT_HEADS;
    float* exb   = ws + off; off += (size_t)tot * GAT_HEADS; // 26.4 MB
    float* agg   = ws + off; off += (size_t)n * GAT_HC;      // 25.6 MB
    float* fmbm  = ws + off; off += 2 * GAT_OUT;
    (void)ws_size; (void)n_in; (void)out_size;

    // Init accumulators (ws is poisoned; must be re-initialized every call)
    {
        long long na = (long long)n * GAT_HEADS;
        long long ng = (long long)n * GAT_HC;
        fill_kernel<<<(unsigned)((na + 255) / 256), 256, 0, stream>>>(amax, -INFINITY, na);
        fill_kernel<<<(unsigned)((na + 255) / 256), 256, 0, stream>>>(denom, 0.0f, na);
        fill_kernel<<<(unsigned)((ng + 255) / 256), 256, 0, stream>>>(agg, 0.0f, ng);
    }

    // 1) xh = x @ W (f32 WMMA)
    gemm_xw_wmma_kernel<<<(n + 15) / 16, 256, 0, stream>>>(x, W, xh, n);

    // 2) attention logits per (node, head)
    attn_scores_kernel<<<(n * GAT_HEADS + 255) / 256, 256, 0, stream>>>(
        xh, att_src, att_dst, a_src, a_dst, n);

    // 3) segment softmax passes over edges (+self loops)
    edge_alpha_max_kernel<<<(tot + 255) / 256, 256, 0, stream>>>(
        src, dst, a_src, a_dst, exb, amax, nE, n);
    edge_exp_sum_kernel<<<(tot + 255) / 256, 256, 0, stream>>>(
        src, dst, exb, amax, denom, nE, n);

    // 4) weighted message scatter (dominant pass)
    edge_scatter_kernel<<<(tot + 1) / 2, 256, 0, stream>>>(
        src, dst, xh, exb, denom, agg, nE, n);

    // 5) FiLM modulators + finalize
    film_kernel<<<1, 32, 0, stream>>>(g, Wf, bf, Wb, bb, fmbm);
    finalize_kernel<<<(n * GAT_OUT + 255) / 256, 256, 0, stream>>>(
        agg, gat_bias, fmbm, (float*)d_out, n);
}